// VectorQuantizer_68375879352379
// MI455X (gfx1250) — compile-verified
//
#include <hip/hip_runtime.h>

typedef float v2f __attribute__((ext_vector_type(2)));
typedef float v8f __attribute__((ext_vector_type(8)));

#define HW 4096          // H*W
#define DDIM 64          // D
#define KCODES 512       // K
#define LDS_STRIDE 66    // 64 + 2 pad: conflict-free banks, keeps 8B alignment

// ---------------------------------------------------------------------------
// ws layout (floats): ws[0] = loss accumulator, ws[16 .. 16+K) = ||c_k||^2
// ---------------------------------------------------------------------------
__global__ void vq_prep(const float* __restrict__ cb, float* __restrict__ ws) {
    int k = threadIdx.x + blockIdx.x * blockDim.x;
    if (k == 0) ws[0] = 0.0f;
    if (k < KCODES) {
        const float4* row = (const float4*)(cb + k * DDIM);
        float s = 0.0f;
#pragma unroll
        for (int i = 0; i < DDIM / 4; ++i) {
            float4 v = row[i];
            s += v.x * v.x + v.y * v.y + v.z * v.z + v.w * v.w;
        }
        ws[16 + k] = s;
    }
}

// ---------------------------------------------------------------------------
// Main kernel: one wave handles 32 tokens (two 16-wide column tiles) vs all
// 512 codes. Per 16-code tile: A = 16 codes x 4 dims (reused twice),
// B{0,1} = 4 dims x 16 tokens, two independent f32 WMMA accumulation chains.
// dist = ||c||^2 - 2*dot  (||z||^2 constant per token -> dropped for argmin)
// ---------------------------------------------------------------------------
__global__ __launch_bounds__(128) void vq_main(const float* __restrict__ z,
                                               const float* __restrict__ cb,
                                               float* __restrict__ ws,
                                               float* __restrict__ out_q,
                                               float* __restrict__ out_idx) {
    __shared__ float lds[4][2][16 * LDS_STRIDE];

    const int lane = threadIdx.x & 31;
    const int wave = threadIdx.x >> 5;
    const int token_base = (blockIdx.x * 4 + wave) * 32;
    const int bb = token_base >> 12;        // batch index (HW = 4096)
    const int s0 = token_base & (HW - 1);   // spatial offset within batch

    // Stage 32 tokens x 64 dims of z into LDS (z layout is (B, D, H, W)).
    const float* zb = z + (size_t)bb * (DDIM * HW) + s0;
#pragma unroll
    for (int j = 0; j < 64; ++j) {
        int flat = j * 32 + lane;           // 0..2047
        int ti = flat & 31;                 // token within the 32-token group
        int d  = flat >> 5;                 // dim
        lds[wave][ti >> 4][(ti & 15) * LDS_STRIDE + d] = zb[(size_t)d * HW + ti];
    }
    __syncthreads();

    const int hi = lane >> 4;               // 0: low half-lane, 1: high
    const int ln = lane & 15;

    float best0 = 3.4e38f, best1 = 3.4e38f;
    int   bidx0 = 0,       bidx1 = 0;

    const float* lz0 = &lds[wave][0][ln * LDS_STRIDE];
    const float* lz1 = &lds[wave][1][ln * LDS_STRIDE];
    const float* cn  = ws + 16;

    for (int tile = 0; tile < KCODES / 16; ++tile) {
        v8f acc0 = {0.f, 0.f, 0.f, 0.f, 0.f, 0.f, 0.f, 0.f};
        v8f acc1 = {0.f, 0.f, 0.f, 0.f, 0.f, 0.f, 0.f, 0.f};
        // A-fragment base: code row (tile*16 + ln), dims hi*2 + {0,1} per step
        const float* arow = cb + (size_t)(tile * 16 + ln) * DDIM + hi * 2;
#pragma unroll
        for (int kk = 0; kk < 16; ++kk) {
            v2f a = *(const v2f*)(arow + kk * 4);
            const float* bp0 = lz0 + kk * 4 + hi * 2;
            const float* bp1 = lz1 + kk * 4 + hi * 2;
            v2f b0; b0.x = bp0[0]; b0.y = bp0[1];
            v2f b1; b1.x = bp1[0]; b1.y = bp1[1];
            acc0 = __builtin_amdgcn_wmma_f32_16x16x4_f32(
                false, a, false, b0, (short)0, acc0, false, false);
            acc1 = __builtin_amdgcn_wmma_f32_16x16x4_f32(
                false, a, false, b1, (short)0, acc1, false, false);
        }
        // C VGPR j holds code M = j + 8*hi for this lane's column (token ln)
#pragma unroll
        for (int j = 0; j < 8; ++j) {
            int code = tile * 16 + hi * 8 + j;
            float cnv = cn[code];
            float dv0 = fmaf(-2.0f, acc0[j], cnv);
            float dv1 = fmaf(-2.0f, acc1[j], cnv);
            if (dv0 < best0) { best0 = dv0; bidx0 = code; }
            if (dv1 < best1) { best1 = dv1; bidx1 = code; }
        }
    }

    // Merge the two half-lanes that share each token column (lane ^ 16).
    {
        float ob = __shfl_xor(best0, 16, 32);
        int   oi = __shfl_xor(bidx0, 16, 32);
        if (ob < best0 || (ob == best0 && oi < bidx0)) { best0 = ob; bidx0 = oi; }
        ob = __shfl_xor(best1, 16, 32);
        oi = __shfl_xor(bidx1, 16, 32);
        if (ob < best1 || (ob == best1 && oi < bidx1)) { best1 = ob; bidx1 = oi; }
    }

    // Write quantized output (B, D, H, W) + accumulate loss for both subtiles.
    float lpart = 0.0f;
#pragma unroll
    for (int u = 0; u < 2; ++u) {
        const int   bidx = u ? bidx1 : bidx0;
        const float* lz  = u ? lz1 : lz0;
        const int   t    = token_base + u * 16 + ln;   // flat token (B,H,W order)
        const int   ss   = s0 + u * 16 + ln;
        float* oq = out_q + (size_t)bb * (DDIM * HW) + ss;
        const float4* crow = (const float4*)(cb + (size_t)bidx * DDIM + hi * 32);
#pragma unroll
        for (int c4 = 0; c4 < 8; ++c4) {    // each half-lane writes 32 dims
            float4 cv = crow[c4];
            int d = hi * 32 + c4 * 4;
            float z0 = lz[d + 0], z1 = lz[d + 1], z2 = lz[d + 2], z3 = lz[d + 3];
            oq[(size_t)(d + 0) * HW] = cv.x;
            oq[(size_t)(d + 1) * HW] = cv.y;
            oq[(size_t)(d + 2) * HW] = cv.z;
            oq[(size_t)(d + 3) * HW] = cv.w;
            float e0 = cv.x - z0, e1 = cv.y - z1, e2 = cv.z - z2, e3 = cv.w - z3;
            lpart += e0 * e0 + e1 * e1 + e2 * e2 + e3 * e3;
        }
        if (hi == 0) out_idx[t] = (float)bidx;
    }

    // Wave-level reduction, one atomic per wave.
#pragma unroll
    for (int off = 16; off >= 1; off >>= 1)
        lpart += __shfl_xor(lpart, off, 32);
    if (lane == 0) atomicAdd(ws, lpart);
}

// loss = codebook_loss + 0.25*commitment = 1.25 * mean((q - z)^2) over N*D
__global__ void vq_finish(const float* __restrict__ ws, float* __restrict__ out_loss) {
    if (threadIdx.x == 0)
        out_loss[0] = 1.25f * ws[0] / 8388608.0f;   // N*D = 131072*64
}

extern "C" void kernel_launch(void* const* d_in, const int* in_sizes, int n_in,
                              void* d_out, int out_size, void* d_ws, size_t ws_size,
                              hipStream_t stream) {
    const float* z  = (const float*)d_in[0];    // (32, 64, 64, 64)
    const float* cb = (const float*)d_in[1];    // (512, 64)
    float* ws  = (float*)d_ws;
    float* out = (float*)d_out;

    float* out_q    = out;                 // 8388608 floats, (B, D, H, W)
    float* out_loss = out + 8388608;       // 1 float
    float* out_idx  = out + 8388609;       // 131072 indices stored as float

    vq_prep  <<<2, 256, 0, stream>>>(cb, ws);
    vq_main  <<<1024, 128, 0, stream>>>(z, cb, ws, out_q, out_idx);
    vq_finish<<<1, 32, 0, stream>>>(ws, out_loss);
}